// GlobalSemanticModule_41601053230005
// MI455X (gfx1250) — compile-verified
//
#include <hip/hip_runtime.h>

typedef float v2f __attribute__((ext_vector_type(2)));
typedef float v8f __attribute__((ext_vector_type(8)));

#define DIM 64
#define RELS 16
#define LDS_STRIDE 66  // 16x64 A tile padded; keeps 8B alignment for paired reads

// ---------------------------------------------------------------------------
// Y[r][n][:] = x[n][:] @ W[r]   (dense, all relations)
// grid.x = n/16 node tiles, grid.y = 2 (relations 0..7 / 8..15)
// block = 256 threads = 8 waves; wave w handles relation blockIdx.y*8 + w.
// WMMA f32 16x16x4:
//   A 16x4 layout: lane M = lane&15; lanes<16 hold K={0,1}, lanes>=16 K={2,3}
//   B 4x16 layout (mirrored): lane N = lane&15; lanes<16 rows {0,1}, >=16 {2,3}
//   C/D 16x16:   VGPR v: lanes<16 -> row v, lanes>=16 -> row v+8; col = lane&15
// ---------------------------------------------------------------------------
__global__ __launch_bounds__(256) void rgcn_gemm_kernel(
    const float* __restrict__ x, const float* __restrict__ w,
    float* __restrict__ y, int num_nodes)
{
    __shared__ float As[16 * LDS_STRIDE];
    const int tid  = threadIdx.x;
    const int base = blockIdx.x * 16;

    // Stage A tile (16 nodes x 64 feats) into LDS, one float4 per thread.
    {
        const int row = tid >> 4;         // 0..15
        const int col = (tid & 15) * 4;   // 0..60
        const float4 v = *(const float4*)(x + (size_t)(base + row) * DIM + col);
        float* dst = As + row * LDS_STRIDE + col;
        dst[0] = v.x; dst[1] = v.y; dst[2] = v.z; dst[3] = v.w;
    }
    __syncthreads();

    const int wave = tid >> 5;                 // 0..7
    const int lane = tid & 31;
    const int r    = blockIdx.y * 8 + wave;    // relation id 0..15
    const int m    = lane & 15;                // M for A, N for B, col for C
    const int koff = (lane >> 4) * 2;          // 0 (lanes 0..15) or 2 (16..31)

    const float* wr = w + (size_t)r * DIM * DIM;

    v8f acc[4] = {};  // 4 N-tiles of 16 cols each

#pragma unroll
    for (int k = 0; k < 16; ++k) {
        const int kb = k * 4 + koff;
        v2f a;
        a.x = As[m * LDS_STRIDE + kb];
        a.y = As[m * LDS_STRIDE + kb + 1];
#pragma unroll
        for (int j = 0; j < 4; ++j) {
            v2f b;
            b.x = wr[(size_t)kb * DIM + j * 16 + m];
            b.y = wr[(size_t)(kb + 1) * DIM + j * 16 + m];
            acc[j] = __builtin_amdgcn_wmma_f32_16x16x4_f32(
                false, a, false, b, (short)0, acc[j], false, false);
        }
    }

    float* yr = y + (size_t)r * num_nodes * DIM;
    const int rowadd = (lane >> 4) * 8;
#pragma unroll
    for (int j = 0; j < 4; ++j) {
        const int col = j * 16 + m;
#pragma unroll
        for (int v = 0; v < 8; ++v) {
            yr[(size_t)(base + v + rowadd) * DIM + col] = acc[j][v];
        }
    }
}

// ---------------------------------------------------------------------------
// aggr[dst] += Y[edge_type][src] / max(path_length, eps)
// one thread per (edge, 4-float chunk); self-loops implicit for e >= num_edges
// ---------------------------------------------------------------------------
__global__ __launch_bounds__(256) void rgcn_scatter_kernel(
    const float* __restrict__ y, const int* __restrict__ edge_index,
    const int* __restrict__ edge_type, const float* __restrict__ path_length,
    float* __restrict__ aggr, int num_nodes, int num_edges)
{
    const int total = (num_edges + num_nodes) * 16;
    const int idx = blockIdx.x * blockDim.x + threadIdx.x;
    if (idx >= total) return;
    const int e = idx >> 4;
    const int q = (idx & 15) * 4;
    int s, d;
    if (e < num_edges) { s = edge_index[e]; d = edge_index[num_edges + e]; }
    else               { s = e - num_edges; d = s; }
    const int r = edge_type[e];
    float pl = path_length[e];
    pl = pl < 1e-9f ? 1e-9f : pl;
    const float inv = 1.0f / pl;
    const float4 mv = *(const float4*)(y + ((size_t)r * num_nodes + s) * DIM + q);
    float* out = aggr + (size_t)d * DIM + q;
    atomicAdd(out + 0, mv.x * inv);
    atomicAdd(out + 1, mv.y * inv);
    atomicAdd(out + 2, mv.z * inv);
    atomicAdd(out + 3, mv.w * inv);
}

// ---------------------------------------------------------------------------
__global__ __launch_bounds__(256) void bias_relu_kernel(
    const float* __restrict__ aggr, const float* __restrict__ bias,
    float* __restrict__ out, int count)
{
    const int i = blockIdx.x * blockDim.x + threadIdx.x;
    if (i >= count) return;
    const float v = aggr[i] + bias[i & (DIM - 1)];
    out[i] = v > 0.f ? v : 0.f;
}

__global__ __launch_bounds__(256) void zero_kernel(float* __restrict__ p, int count)
{
    const int i = blockIdx.x * blockDim.x + threadIdx.x;
    if (i < count) p[i] = 0.f;
}

// ---------------------------------------------------------------------------
extern "C" void kernel_launch(void* const* d_in, const int* in_sizes, int n_in,
                              void* d_out, int out_size, void* d_ws, size_t ws_size,
                              hipStream_t stream)
{
    const float* x           = (const float*)d_in[0];
    const int*   edge_index  = (const int*)  d_in[1];
    const int*   edge_type   = (const int*)  d_in[2];
    const float* path_length = (const float*)d_in[3];
    const float* w1          = (const float*)d_in[4];
    const float* b1          = (const float*)d_in[5];
    const float* w2          = (const float*)d_in[6];
    const float* b2          = (const float*)d_in[7];
    float*       out         = (float*)d_out;

    const int n = in_sizes[0] / DIM;   // num nodes
    const int e = in_sizes[1] / 2;     // num edges (pre self-loop)

    // workspace layout: Y (R*n*D) | aggr (n*D) | x1 (n*D)
    float* y    = (float*)d_ws;
    float* aggr = y + (size_t)RELS * n * DIM;
    float* x1   = aggr + (size_t)n * DIM;

    const int  ntiles = (n + 15) / 16;
    const dim3 ggrid(ntiles, 2);
    const int  nd       = n * DIM;
    const int  nblocks  = (nd + 255) / 256;
    const int  stotal   = (e + n) * 16;
    const int  sblocks  = (stotal + 255) / 256;

    // ---- layer 1 ----
    zero_kernel        <<<nblocks, 256, 0, stream>>>(aggr, nd);
    rgcn_gemm_kernel   <<<ggrid,   256, 0, stream>>>(x, w1, y, n);
    rgcn_scatter_kernel<<<sblocks, 256, 0, stream>>>(y, edge_index, edge_type,
                                                     path_length, aggr, n, e);
    bias_relu_kernel   <<<nblocks, 256, 0, stream>>>(aggr, b1, x1, nd);

    // ---- layer 2 ----
    zero_kernel        <<<nblocks, 256, 0, stream>>>(aggr, nd);
    rgcn_gemm_kernel   <<<ggrid,   256, 0, stream>>>(x1, w2, y, n);
    rgcn_scatter_kernel<<<sblocks, 256, 0, stream>>>(y, edge_index, edge_type,
                                                     path_length, aggr, n, e);
    bias_relu_kernel   <<<nblocks, 256, 0, stream>>>(aggr, b2, out, nd);
}